// DBGNN_16338055594020
// MI455X (gfx1250) — compile-verified
//
#include <hip/hip_runtime.h>

typedef __attribute__((ext_vector_type(2))) float v2f;
typedef __attribute__((ext_vector_type(8))) float v8f;

#define NA    100000
#define NT    25000
#define CCOLS 8
#define VVOC  1000
#define DDIM  64
#define KP    512    // C*D
#define KPS   516    // padded LDS stride (516 % 64 = 4 -> conflict-free)
#define PP    128    // proj dim
#define PPS   132    // padded LDS stride (132 % 64 = 4 -> conflict-free)
#define OUTC  16

static __device__ __forceinline__ v8f wmma_f32(v2f a, v2f b, v8f c) {
  return __builtin_amdgcn_wmma_f32_16x16x4_f32(false, a, false, b, (short)0, c,
                                               false, false);
}

// ---------- elementwise helpers ----------
__global__ void fill_kernel(float* __restrict__ p, float v, int n) {
  int i = blockIdx.x * blockDim.x + threadIdx.x;
  if (i < n) p[i] = v;
}

__global__ void combine_kernel(const float* __restrict__ o1,
                               const float* __restrict__ o2,
                               float* __restrict__ h, int n) {
  int i = blockIdx.x * blockDim.x + threadIdx.x;
  if (i < n) h[i] = 0.5f * (o1[i] + o2[i]);
}

__global__ void divide_kernel(float* __restrict__ agg,
                              const float* __restrict__ cnt, int nrows) {
  int i = blockIdx.x * blockDim.x + threadIdx.x;
  if (i < nrows * PP) {
    int r = i >> 7;
    float c = cnt[r];
    agg[i] = agg[i] / fmaxf(c, 1.0f);
  }
}

// ---------- edge scatter-add (one wave per edge, float4 per lane) ----------
__global__ __launch_bounds__(256) void scatter_add_kernel(
    const float* __restrict__ x, const int* __restrict__ src,
    const int* __restrict__ dst, int n_edges, float* __restrict__ agg,
    float* __restrict__ cnt) {
  int e = blockIdx.x * 8 + (threadIdx.x >> 5);
  if (e >= n_edges) return;
  int lane = threadIdx.x & 31;
  int s = src[e], d = dst[e];
  const float* xr = x + (size_t)s * PP;
  float* ar = agg + (size_t)d * PP;
#pragma unroll
  for (int j = 0; j < 4; ++j) {
    atomicAdd(&ar[lane * 4 + j], xr[lane * 4 + j]);
  }
  if (lane == 0) atomicAdd(&cnt[d], 1.0f);
}

// ---------- embedding gather + projection: h_a = emb @ Wp + bp ----------
__global__ __launch_bounds__(256) void embed_project_kernel(
    const int* __restrict__ x_a, const float* __restrict__ table,  // [C,V,D]
    const float* __restrict__ Wp,                                  // [512,128]
    const float* __restrict__ bp, float* __restrict__ h_a, int n_rows) {
  __shared__ float Atile[16 * KPS];  // ~33 KB, padded stride
  const int row0 = blockIdx.x * 16;
  const int tid = threadIdx.x;
  const bool full = (row0 + 16) <= n_rows;
  for (int e = tid; e < 16 * KP; e += 256) {
    int r = e >> 9;   // /512
    int k = e & 511;
    int c = k >> 6;   // /64
    int d = k & 63;
    float v = 0.0f;
    int row = row0 + r;
    if (full || row < n_rows) {
      int idx = x_a[row * CCOLS + c];
      v = table[(c * VVOC + idx) * DDIM + d];
    }
    Atile[r * KPS + k] = v;
  }
  __syncthreads();
  const int wave = tid >> 5;   // 8 waves -> 8 N-tiles of 16
  const int lane = tid & 31;
  const int n0 = wave * 16;
  const int half = lane >> 4;  // 0: K+{0,1}, 1: K+{2,3}
  const int l16 = lane & 15;
  v8f acc = {};
  for (int k = 0; k < KP; k += 4) {
    v2f a, b;
    a.x = Atile[l16 * KPS + k + 2 * half];
    a.y = Atile[l16 * KPS + k + 2 * half + 1];
    b.x = Wp[(k + 2 * half) * PP + n0 + l16];
    b.y = Wp[(k + 2 * half + 1) * PP + n0 + l16];
    acc = wmma_f32(a, b, acc);
  }
  float bias = bp[n0 + l16];
  if (full) {
#pragma unroll
    for (int r = 0; r < 8; ++r)
      h_a[(size_t)(row0 + r + 8 * half) * PP + n0 + l16] = acc[r] + bias;
  } else {
#pragma unroll
    for (int r = 0; r < 8; ++r) {
      int row = row0 + r + 8 * half;
      if (row < n_rows) h_a[(size_t)row * PP + n0 + l16] = acc[r] + bias;
    }
  }
}

// ---------- fused SAGE: out = agg@Wl + bl + h@Wr  (32 rows / block) ----------
__global__ __launch_bounds__(256) void sage_kernel(
    const float* __restrict__ agg, const float* __restrict__ h,
    const float* __restrict__ Wl, const float* __restrict__ bl,
    const float* __restrict__ Wr, float* __restrict__ out, int n_rows) {
  __shared__ float Aagg[32 * PPS];  // ~16.9 KB each, padded stride
  __shared__ float Ah[32 * PPS];
  const int row0 = blockIdx.x * 32;
  const int tid = threadIdx.x;
  const bool full = (row0 + 32) <= n_rows;
  if (full) {
    for (int e = tid; e < 32 * PP; e += 256) {
      int r = e >> 7;
      int k = e & 127;
      size_t g = (size_t)(row0 + r) * PP + k;
      Aagg[r * PPS + k] = agg[g];
      Ah[r * PPS + k] = h[g];
    }
  } else {
    for (int e = tid; e < 32 * PP; e += 256) {
      int r = e >> 7;
      int k = e & 127;
      int row = row0 + r;
      float va = 0.0f, vh = 0.0f;
      if (row < n_rows) {
        va = agg[(size_t)row * PP + k];
        vh = h[(size_t)row * PP + k];
      }
      Aagg[r * PPS + k] = va;
      Ah[r * PPS + k] = vh;
    }
  }
  __syncthreads();
  const int wave = tid >> 5;
  const int lane = tid & 31;
  const int n0 = wave * 16;
  const int half = lane >> 4;
  const int l16 = lane & 15;
  v8f acc0 = {}, acc1 = {};  // row-tiles [row0, row0+16) and [row0+16, row0+32)
  for (int k = 0; k < PP; k += 4) {  // agg @ Wl : one B-frag feeds 2 WMMAs
    v2f b, a0, a1;
    b.x = Wl[(k + 2 * half) * PP + n0 + l16];
    b.y = Wl[(k + 2 * half + 1) * PP + n0 + l16];
    a0.x = Aagg[l16 * PPS + k + 2 * half];
    a0.y = Aagg[l16 * PPS + k + 2 * half + 1];
    a1.x = Aagg[(16 + l16) * PPS + k + 2 * half];
    a1.y = Aagg[(16 + l16) * PPS + k + 2 * half + 1];
    acc0 = wmma_f32(a0, b, acc0);
    acc1 = wmma_f32(a1, b, acc1);
  }
  for (int k = 0; k < PP; k += 4) {  // + h @ Wr (same accumulators)
    v2f b, a0, a1;
    b.x = Wr[(k + 2 * half) * PP + n0 + l16];
    b.y = Wr[(k + 2 * half + 1) * PP + n0 + l16];
    a0.x = Ah[l16 * PPS + k + 2 * half];
    a0.y = Ah[l16 * PPS + k + 2 * half + 1];
    a1.x = Ah[(16 + l16) * PPS + k + 2 * half];
    a1.y = Ah[(16 + l16) * PPS + k + 2 * half + 1];
    acc0 = wmma_f32(a0, b, acc0);
    acc1 = wmma_f32(a1, b, acc1);
  }
  float bias = bl[n0 + l16];
  if (full) {
#pragma unroll
    for (int r = 0; r < 8; ++r) {
      out[(size_t)(row0 + r + 8 * half) * PP + n0 + l16] = acc0[r] + bias;
      out[(size_t)(row0 + 16 + r + 8 * half) * PP + n0 + l16] =
          acc1[r] + bias;
    }
  } else {
#pragma unroll
    for (int r = 0; r < 8; ++r) {
      int rowA = row0 + r + 8 * half;
      if (rowA < n_rows) out[(size_t)rowA * PP + n0 + l16] = acc0[r] + bias;
      int rowB = row0 + 16 + r + 8 * half;
      if (rowB < n_rows) out[(size_t)rowB * PP + n0 + l16] = acc1[r] + bias;
    }
  }
}

// ---------- final: softmax(h_t @ Wout + bout) ----------
__global__ __launch_bounds__(256) void out_softmax_kernel(
    const float* __restrict__ ht, const float* __restrict__ Wout,  // [128,16]
    const float* __restrict__ bout, float* __restrict__ out, int n_rows) {
  const int wave = threadIdx.x >> 5;
  const int lane = threadIdx.x & 31;
  const int row0 = blockIdx.x * 128 + wave * 16;
  if (row0 >= n_rows) return;  // wave-uniform
  const bool full = (row0 + 16) <= n_rows;
  const int half = lane >> 4;
  const int l16 = lane & 15;
  v8f acc = {};
  const int rA = row0 + l16;
  for (int k = 0; k < PP; k += 4) {
    v2f a, b;
    float a0 = 0.0f, a1 = 0.0f;
    if (full || rA < n_rows) {
      a0 = ht[(size_t)rA * PP + k + 2 * half];
      a1 = ht[(size_t)rA * PP + k + 2 * half + 1];
    }
    a.x = a0;
    a.y = a1;
    b.x = Wout[(k + 2 * half) * OUTC + l16];
    b.y = Wout[(k + 2 * half + 1) * OUTC + l16];
    acc = wmma_f32(a, b, acc);
  }
  float bias = bout[l16];
#pragma unroll
  for (int r = 0; r < 8; ++r) {
    float v = acc[r] + bias;
    float m = v;  // max over the 16 lanes of this half (xor<16 stays in half)
    for (int off = 1; off < 16; off <<= 1) m = fmaxf(m, __shfl_xor(m, off, 32));
    float e = expf(v - m);
    float s = e;
    for (int off = 1; off < 16; off <<= 1) s += __shfl_xor(s, off, 32);
    int row = row0 + r + 8 * half;
    if (full || row < n_rows) out[(size_t)row * OUTC + l16] = e / s;
  }
}

// ============================ host launcher ============================
extern "C" void kernel_launch(void* const* d_in, const int* in_sizes, int n_in,
                              void* d_out, int out_size, void* d_ws,
                              size_t ws_size, hipStream_t stream) {
  const int* x_a = (const int*)d_in[0];
  const int* aa_src = (const int*)d_in[1];
  const int* aa_dst = (const int*)d_in[2];
  const int* at_src = (const int*)d_in[3];
  const int* at_dst = (const int*)d_in[4];
  const int* ta_src = (const int*)d_in[5];
  const int* ta_dst = (const int*)d_in[6];
  const float* table = (const float*)d_in[8];
  const float* Wp = (const float*)d_in[9];
  const float* bp = (const float*)d_in[10];
  const float* Wl = (const float*)d_in[11];
  const float* bl = (const float*)d_in[12];
  const float* Wr = (const float*)d_in[13];
  const float* Wout = (const float*)d_in[14];
  const float* bout = (const float*)d_in[15];
  const int E_AA = in_sizes[1];
  const int E_AT = in_sizes[3];
  const int E_TA = in_sizes[5];

  float* ws = (float*)d_ws;
  size_t off = 0;
  float* h_a = ws + off;  off += (size_t)NA * PP;
  float* ht0 = ws + off;  off += (size_t)NT * PP;
  float* ht1 = ws + off;  off += (size_t)NT * PP;
  float* agg = ws + off;  off += (size_t)NA * PP;
  float* cnt = ws + off;  off += (size_t)NA;
  float* o1  = ws + off;  off += (size_t)NA * PP;
  float* o2  = ws + off;  off += (size_t)NA * PP;
  (void)ws_size; (void)n_in; (void)out_size;

  auto grid1 = [](int n) { return dim3((n + 255) / 256); };

  // h_t init = ones; h_a = embed+project
  fill_kernel<<<grid1(NT * PP), 256, 0, stream>>>(ht0, 1.0f, NT * PP);
  embed_project_kernel<<<dim3((NA + 15) / 16), 256, 0, stream>>>(
      x_a, table, Wp, bp, h_a, NA);

  float* htc = ht0;
  for (int l = 0; l < 2; ++l) {
    float* htn = (l == 0) ? ht1 : ht0;
    const float* WL = Wl + (size_t)(l * 3) * PP * PP;
    const float* BL = bl + (size_t)(l * 3) * PP;
    const float* WR = Wr + (size_t)(l * 3) * PP * PP;

    // o_at -> htn   (dst = target nodes)
    fill_kernel<<<grid1(NT * PP), 256, 0, stream>>>(agg, 0.0f, NT * PP);
    fill_kernel<<<grid1(NT), 256, 0, stream>>>(cnt, 0.0f, NT);
    scatter_add_kernel<<<dim3((E_AT + 7) / 8), 256, 0, stream>>>(
        h_a, at_src, at_dst, E_AT, agg, cnt);
    divide_kernel<<<grid1(NT * PP), 256, 0, stream>>>(agg, cnt, NT);
    sage_kernel<<<dim3((NT + 31) / 32), 256, 0, stream>>>(
        agg, htc, WL + (size_t)1 * PP * PP, BL + 1 * PP,
        WR + (size_t)1 * PP * PP, htn, NT);

    // o_aa -> o1
    fill_kernel<<<grid1(NA * PP), 256, 0, stream>>>(agg, 0.0f, NA * PP);
    fill_kernel<<<grid1(NA), 256, 0, stream>>>(cnt, 0.0f, NA);
    scatter_add_kernel<<<dim3((E_AA + 7) / 8), 256, 0, stream>>>(
        h_a, aa_src, aa_dst, E_AA, agg, cnt);
    divide_kernel<<<grid1(NA * PP), 256, 0, stream>>>(agg, cnt, NA);
    sage_kernel<<<dim3((NA + 31) / 32), 256, 0, stream>>>(
        agg, h_a, WL, BL, WR, o1, NA);

    // o_ta -> o2
    fill_kernel<<<grid1(NA * PP), 256, 0, stream>>>(agg, 0.0f, NA * PP);
    fill_kernel<<<grid1(NA), 256, 0, stream>>>(cnt, 0.0f, NA);
    scatter_add_kernel<<<dim3((E_TA + 7) / 8), 256, 0, stream>>>(
        htc, ta_src, ta_dst, E_TA, agg, cnt);
    divide_kernel<<<grid1(NA * PP), 256, 0, stream>>>(agg, cnt, NA);
    sage_kernel<<<dim3((NA + 31) / 32), 256, 0, stream>>>(
        agg, h_a, WL + (size_t)2 * PP * PP, BL + 2 * PP,
        WR + (size_t)2 * PP * PP, o2, NA);

    // h_a = 0.5*(o_aa + o_ta); h_t = o_at (pointer swap)
    combine_kernel<<<grid1(NA * PP), 256, 0, stream>>>(o1, o2, h_a, NA * PP);
    htc = htn;
  }

  out_softmax_kernel<<<dim3((NT + 127) / 128), 256, 0, stream>>>(
      htc, Wout, bout, (float*)d_out, NT);
}